// ExphormerAttention_79774722555995
// MI455X (gfx1250) — compile-verified
//
#include <hip/hip_runtime.h>
#include <hip/hip_bf16.h>

typedef __attribute__((ext_vector_type(2))) float v2f;
typedef __attribute__((ext_vector_type(4))) float v4f;
typedef __attribute__((ext_vector_type(8))) float v8f;

#define NN 100000
#define NEDGE 1250000
#define DD 64
#define HH 8
#define DHH 8

// ---------------------------------------------------------------------------
// Kernel 0: zero the output accumulator and the Z accumulator
// ---------------------------------------------------------------------------
__global__ void exph_zero(float* __restrict__ out, float* __restrict__ Z) {
    int i = blockIdx.x * blockDim.x + threadIdx.x;
    if (i < NN * DD) out[i] = 0.0f;
    if (i < NN * HH) Z[i] = 0.0f;
}

// ---------------------------------------------------------------------------
// Kernel 1: Q/K/V projections via V_WMMA_F32_16X16X4_F32
// One wave per 16-row tile of x. x tile staged in LDS (padded stride 68).
// ---------------------------------------------------------------------------
#define PW 4  // waves per block
__global__ __launch_bounds__(128) void exph_proj(
    const float* __restrict__ x,
    const float* __restrict__ WQ, const float* __restrict__ WK,
    const float* __restrict__ WV,
    float* __restrict__ Q, float* __restrict__ K, float* __restrict__ V) {
    __shared__ float xs[PW][16 * 68];

    const int wave = threadIdx.x >> 5;
    const int lane = threadIdx.x & 31;
    const int m    = lane & 15;
    const int hi   = lane >> 4;

    const int tile = blockIdx.x * PW + wave;
    if (tile * 16 >= NN) return;
    const int row0 = tile * 16;

    // Stage 16x64 tile of x (contiguous 1024 floats), coalesced float4 loads,
    // write into padded-row LDS.
    #pragma unroll
    for (int i = 0; i < 8; ++i) {
        int j   = i * 128 + lane * 4;    // element id within the tile
        int r   = j >> 6;
        int c   = j & 63;
        *(v4f*)&xs[wave][r * 68 + c] = *(const v4f*)&x[(size_t)row0 * 64 + j];
    }
    asm volatile("s_wait_dscnt 0x0" ::: "memory");

    // A fragments for all 16 k-steps (16x4 f32 A layout: M=lane%16,
    // VGPR0: K = kb*4 + 2*hi, VGPR1: +1)
    v2f a[16];
    #pragma unroll
    for (int kb = 0; kb < 16; ++kb)
        a[kb] = *(const v2f*)&xs[wave][m * 68 + kb * 4 + 2 * hi];

    const float* Ws[3]  = {WQ, WK, WV};
    float*       Os[3]  = {Q, K, V};

    for (int w = 0; w < 3; ++w) {
        const float* W = Ws[w];
        float*       O = Os[w];
        #pragma unroll
        for (int c = 0; c < 4; ++c) {
            v8f acc = {};
            #pragma unroll
            for (int kb = 0; kb < 16; ++kb) {
                // B 4x16 f32: VGPR0 rows K = kb*4 + 2*hi, VGPR1 row +1,
                // N = lane%16
                int r0 = kb * 4 + 2 * hi;
                v2f b;
                b.x = W[r0 * 64 + c * 16 + m];
                b.y = W[(r0 + 1) * 64 + c * 16 + m];
                acc = __builtin_amdgcn_wmma_f32_16x16x4_f32(
                    false, a[kb], false, b, (short)0, acc, false, false);
            }
            // C/D layout: VGPR r -> row r (lanes 0-15) / row r+8 (lanes 16-31)
            #pragma unroll
            for (int r = 0; r < 8; ++r) {
                int M = r + hi * 8;
                O[(size_t)(row0 + M) * 64 + c * 16 + m] = acc[r];
            }
        }
    }
}

// ---------------------------------------------------------------------------
// Kernel 2: per-edge phase. Fused Ef = edge_attr @ WE (WMMA), score, exp,
// atomic scatter into out (wV) and Z.
// One wave handles 16 edges. Block = 4 waves = 64 edges.
// ---------------------------------------------------------------------------
#define EW 4
__global__ __launch_bounds__(128) void exph_edge(
    const float* __restrict__ edge_attr, const float* __restrict__ WE,
    const int* __restrict__ ei,
    const float* __restrict__ Qp, const float* __restrict__ Kp,
    const float* __restrict__ Vp,
    float* __restrict__ out, float* __restrict__ Zp) {
    __shared__ float wT[64 * 68];        // WE transposed: wT[col*68 + row]
    __shared__ float eAt[EW][16 * 68];   // staged edge_attr tiles
    __shared__ float ef[EW][16 * 68];    // Ef tiles

    const int tid  = threadIdx.x;
    const int wave = tid >> 5;
    const int lane = tid & 31;
    const int m    = lane & 15;
    const int hi   = lane >> 4;

    // Stage WE transposed (once per block)
    for (int i = 0; i < 32; ++i) {
        int j = i * 128 + tid;           // j in [0,4096)
        int r = j >> 6;                  // k (row of WE)
        int c = j & 63;                  // n (col of WE)
        wT[c * 68 + r] = WE[j];
    }
    __syncthreads();

    const int tileE = blockIdx.x * EW + wave;
    const int NT    = NEDGE / 16;        // 78125, exact
    if (tileE < NT) {
        const int e0 = tileE * 16;

        // Stage 16x64 edge_attr tile (contiguous rows), coalesced.
        #pragma unroll
        for (int i = 0; i < 8; ++i) {
            int j = i * 128 + lane * 4;
            int r = j >> 6;
            int c = j & 63;
            *(v4f*)&eAt[wave][r * 68 + c] =
                *(const v4f*)&edge_attr[(size_t)e0 * 64 + j];
        }
        asm volatile("s_wait_dscnt 0x0" ::: "memory");

        // A fragments (edges x K)
        v2f a[16];
        #pragma unroll
        for (int kb = 0; kb < 16; ++kb)
            a[kb] = *(const v2f*)&eAt[wave][m * 68 + kb * 4 + 2 * hi];

        // Ef = edge_attr_tile @ WE via WMMA, result staged to LDS
        #pragma unroll
        for (int c = 0; c < 4; ++c) {
            v8f acc = {};
            #pragma unroll
            for (int kb = 0; kb < 16; ++kb) {
                // b.x = WE[kb*4+2*hi][c*16+m], b.y = next row -> adjacent in wT
                v2f b = *(const v2f*)&wT[(c * 16 + m) * 68 + kb * 4 + 2 * hi];
                acc = __builtin_amdgcn_wmma_f32_16x16x4_f32(
                    false, a[kb], false, b, (short)0, acc, false, false);
            }
            #pragma unroll
            for (int r = 0; r < 8; ++r) {
                int M = r + hi * 8;
                ef[wave][M * 68 + c * 16 + m] = acc[r];
            }
        }
        asm volatile("s_wait_dscnt 0x0" ::: "memory");

        // Score + scatter: lane handles (edge m2, head h); 4 iterations
        const float inv_sqrt_dh = 0.35355339059327379f;
        #pragma unroll
        for (int t = 0; t < 4; ++t) {
            int p  = t * 32 + lane;
            int m2 = p >> 3;
            int h  = p & 7;
            int e  = e0 + m2;
            int sN = ei[e];              // src
            int dN = ei[NEDGE + e];      // dst

            const v4f K0 = *(const v4f*)&Kp[(size_t)sN * 64 + h * 8];
            const v4f K1 = *(const v4f*)&Kp[(size_t)sN * 64 + h * 8 + 4];
            const v4f Q0 = *(const v4f*)&Qp[(size_t)dN * 64 + h * 8];
            const v4f Q1 = *(const v4f*)&Qp[(size_t)dN * 64 + h * 8 + 4];
            const v4f E0 = *(const v4f*)&ef[wave][m2 * 68 + h * 8];
            const v4f E1 = *(const v4f*)&ef[wave][m2 * 68 + h * 8 + 4];

            float s = K0.x * Q0.x * E0.x + K0.y * Q0.y * E0.y +
                      K0.z * Q0.z * E0.z + K0.w * Q0.w * E0.w +
                      K1.x * Q1.x * E1.x + K1.y * Q1.y * E1.y +
                      K1.z * Q1.z * E1.z + K1.w * Q1.w * E1.w;
            s = s * inv_sqrt_dh;
            s = fminf(5.0f, fmaxf(-5.0f, s));
            const float sc = __expf(s);

            const v4f V0 = *(const v4f*)&Vp[(size_t)sN * 64 + h * 8];
            const v4f V1 = *(const v4f*)&Vp[(size_t)sN * 64 + h * 8 + 4];

            float* ob = out + (size_t)dN * 64 + h * 8;
            atomicAdd(ob + 0, V0.x * sc);
            atomicAdd(ob + 1, V0.y * sc);
            atomicAdd(ob + 2, V0.z * sc);
            atomicAdd(ob + 3, V0.w * sc);
            atomicAdd(ob + 4, V1.x * sc);
            atomicAdd(ob + 5, V1.y * sc);
            atomicAdd(ob + 6, V1.z * sc);
            atomicAdd(ob + 7, V1.w * sc);
            atomicAdd(&Zp[(size_t)dN * 8 + h], sc);
        }
    }
}

// ---------------------------------------------------------------------------
// Kernel 3: normalize out by Z
// ---------------------------------------------------------------------------
__global__ void exph_norm(float* __restrict__ out, const float* __restrict__ Z) {
    int i = blockIdx.x * blockDim.x + threadIdx.x;
    if (i < NN * DD) {
        int n = i >> 6;
        int h = (i & 63) >> 3;
        out[i] = out[i] / (Z[n * 8 + h] + 1e-6f);
    }
}

// ---------------------------------------------------------------------------
extern "C" void kernel_launch(void* const* d_in, const int* in_sizes, int n_in,
                              void* d_out, int out_size, void* d_ws,
                              size_t ws_size, hipStream_t stream) {
    const float* x   = (const float*)d_in[0];
    const float* ea  = (const float*)d_in[1];
    const float* WQ  = (const float*)d_in[2];
    const float* WK  = (const float*)d_in[3];
    const float* WV  = (const float*)d_in[4];
    const float* WE  = (const float*)d_in[5];
    const int*   ei  = (const int*)d_in[6];
    float*       out = (float*)d_out;

    float* ws = (float*)d_ws;
    float* Q  = ws;                       // N*64
    float* K  = ws + (size_t)NN * 64;     // N*64
    float* V  = ws + (size_t)NN * 128;    // N*64
    float* Z  = ws + (size_t)NN * 192;    // N*8   (total 80 MB)

    // 0) zero accumulators (must happen every call: graph replays accumulate)
    exph_zero<<<(NN * DD + 255) / 256, 256, 0, stream>>>(out, Z);

    // 1) Q/K/V projections (6250 tiles of 16 rows, 4 waves/block)
    {
        int tiles  = NN / 16;                 // 6250 exact
        int blocks = (tiles + PW - 1) / PW;   // 1563
        exph_proj<<<blocks, 128, 0, stream>>>(x, WQ, WK, WV, Q, K, V);
    }

    // 2) edge phase (78125 tiles of 16 edges, 4 waves/block)
    {
        int tiles  = NEDGE / 16;              // 78125 exact
        int blocks = (tiles + EW - 1) / EW;   // 19532
        exph_edge<<<blocks, 128, 0, stream>>>(ea, WE, ei, Q, K, V, out, Z);
    }

    // 3) normalize
    exph_norm<<<(NN * DD + 255) / 256, 256, 0, stream>>>(out, Z);
}